// PSOAlignment_56831007261042
// MI455X (gfx1250) — compile-verified
//
#include <hip/hip_runtime.h>
#include <hip/hip_bf16.h>
#include <math.h>

// ---------------- problem constants (match reference setup_inputs) ----------
constexpr int   BATCH = 64;
constexpr int   NPTS  = 16384;   // N == M
constexpr int   SSUB  = 128;     // SUBSAMPLE
constexpr int   NPART = 50;      // N_PARTICLES
constexpr int   NITER = 30;
constexpr float ROT_RANGE   = 0.7853981633974483f;  // 45 deg
constexpr float TRANS_RANGE = 1.0f;

typedef __attribute__((ext_vector_type(2))) float v2f;
typedef __attribute__((ext_vector_type(8))) float v8f;

// ---------------- small deterministic hash RNG ------------------------------
__device__ __forceinline__ unsigned hashu(unsigned x) {
    x ^= x >> 16; x *= 0x7feb352du;
    x ^= x >> 15; x *= 0x846ca68bu;
    x ^= x >> 16; return x;
}
__device__ __forceinline__ float u01(unsigned x) {
    return (float)(hashu(x) >> 8) * (1.0f / 16777216.0f);
}

// fast single-instruction transcendentals (v_sqrt_f32 / v_rcp_f32)
__device__ __forceinline__ float fast_sqrt(float x) { return __builtin_amdgcn_sqrtf(x); }
__device__ __forceinline__ float fast_rcp(float x)  { return __builtin_amdgcn_rcpf(x); }

// ---------------- Rodrigues axis-angle -> R ---------------------------------
__device__ __forceinline__ void rodrigues(float ax, float ay, float az, float R[9]) {
    float th = fast_sqrt(ax * ax + ay * ay + az * az);
    th = fmaxf(th, 1e-8f);
    float inv = fast_rcp(th);
    float ux = ax * inv, uy = ay * inv, uz = az * inv;
    float s = __sinf(th), c = __cosf(th), k = 1.0f - c;
    R[0] = c + k * ux * ux;      R[1] = k * ux * uy - s * uz; R[2] = k * ux * uz + s * uy;
    R[3] = k * ux * uy + s * uz; R[4] = c + k * uy * uy;      R[5] = k * uy * uz - s * ux;
    R[6] = k * ux * uz - s * uy; R[7] = k * uy * uz + s * ux; R[8] = c + k * uz * uz;
}

// ---------------- kernel 1: subsample source/target, precompute |t|^2 -------
__global__ __launch_bounds__(SSUB) void subsample_kernel(
    const float* __restrict__ src, const float* __restrict__ tgt,
    float* __restrict__ ssx, float* __restrict__ ssy, float* __restrict__ ssz,
    float* __restrict__ ttx, float* __restrict__ tty, float* __restrict__ ttz,
    float* __restrict__ ttn)
{
    int b = blockIdx.x, i = threadIdx.x;
    const int stride = NPTS / SSUB;                 // 128: deterministic strided subsample
    size_t gi = ((size_t)b * NPTS + (size_t)i * stride) * 3;
    int o = b * SSUB + i;
    ssx[o] = src[gi + 0]; ssy[o] = src[gi + 1]; ssz[o] = src[gi + 2];
    float x = tgt[gi + 0], y = tgt[gi + 1], z = tgt[gi + 2];
    ttx[o] = x; tty[o] = y; ttz[o] = z; ttn[o] = x * x + y * y + z * z;
}

// ---------------- kernel 2: init particles ----------------------------------
__global__ void init_particles(float* __restrict__ pos, float* __restrict__ vel)
{
    int i = blockIdx.x * blockDim.x + threadIdx.x;
    if (i >= BATCH * NPART) return;
    int p = i % NPART;
    #pragma unroll
    for (int d = 0; d < 6; ++d) {
        float range = (d < 3) ? ROT_RANGE : TRANS_RANGE;
        float up = u01((unsigned)i * 977u + (unsigned)d * 131u + 0x9E3779B9u);
        float uv = u01((unsigned)i * 977u + (unsigned)d * 131u + 0x85EBCA6Bu);
        pos[i * 6 + d] = (p == 0) ? 0.0f : (2.0f * up - 1.0f) * range;
        vel[i * 6 + d] = (2.0f * uv - 1.0f) * range * 0.1f;
    }
}

// ---------------- kernel 3: fitness eval (WMMA chamfer) ---------------------
// One 128-thread block per (batch, particle). 128x128 squared-distance matrix
// as 8x8 tiles of 16x16; one V_WMMA_F32_16X16X4_F32 per tile computes
// d2 = |s|^2 (in C) + (-2s . t) + |t|^2 (K=4 pad trick).
// Mins are taken on d2 (sqrt is monotonic); sqrt applied only at the 128+128
// row/col minima, not per element.
__global__ __launch_bounds__(128) void eval_wmma_kernel(
    const float* __restrict__ ssx, const float* __restrict__ ssy, const float* __restrict__ ssz,
    const float* __restrict__ ttx, const float* __restrict__ tty, const float* __restrict__ ttz,
    const float* __restrict__ ttn,
    const float* __restrict__ pos, float* __restrict__ fit)
{
    __shared__ float sx[SSUB], sy[SSUB], sz[SSUB], sn[SSUB];
    __shared__ float tx[SSUB], ty[SSUB], tz[SSUB], tn[SSUB];
    __shared__ unsigned colMinU[SSUB];     // min of d2 per column (bit-ordered)
    __shared__ float rowMinS[SSUB];        // sqrt(min d2) per row
    __shared__ float red[SSUB];

    const int tid = threadIdx.x;
    const int bp  = blockIdx.x;
    const int b   = bp / NPART;

    // pose -> R,t (uniform per block; computed redundantly, cheap)
    const float* q = pos + (size_t)bp * 6;
    float R[9];
    rodrigues(q[0], q[1], q[2], R);
    const float t0 = q[3], t1 = q[4], t2 = q[5];

    // transform source subsample, stage everything in LDS
    {
        int o = b * SSUB + tid;
        float x = ssx[o], y = ssy[o], z = ssz[o];
        float px = R[0] * x + R[1] * y + R[2] * z + t0;
        float py = R[3] * x + R[4] * y + R[5] * z + t1;
        float pz = R[6] * x + R[7] * y + R[8] * z + t2;
        sx[tid] = px; sy[tid] = py; sz[tid] = pz;
        sn[tid] = px * px + py * py + pz * pz;
        tx[tid] = ttx[o]; ty[tid] = tty[o]; tz[tid] = ttz[o]; tn[tid] = ttn[o];
        colMinU[tid] = 0x7F800000u;   // +inf
    }
    __syncthreads();

    const int wave = tid >> 5;
    const int lane = tid & 31;
    const int half = lane >> 4;       // A/B K-split group
    const int lr   = lane & 15;

    // Preload B fragments for all 8 column tiles (independent of row tile).
    // B (4x16): lanes 0-15 -> rows K0,K1 ; lanes 16-31 -> rows K2,K3
    v2f Bt[8];
    #pragma unroll
    for (int ct = 0; ct < 8; ++ct) {
        const int Ncol = ct * 16 + lr;
        Bt[ct].x = (half == 0) ? tx[Ncol] : tz[Ncol];
        Bt[ct].y = (half == 0) ? ty[Ncol] : tn[Ncol];
    }

    // each wave owns 2 row tiles (32 rows)
    #pragma unroll
    for (int rti = 0; rti < 2; ++rti) {
        const int rowBase = (wave * 2 + rti) * 16;
        const int Midx = rowBase + lr;

        // A (16x4): lanes 0-15 -> K0,K1 ; lanes 16-31 -> K2,K3
        v2f A;
        A.x = (half == 0) ? (-2.0f * sx[Midx]) : (-2.0f * sz[Midx]);
        A.y = (half == 0) ? (-2.0f * sy[Midx]) : 1.0f;

        // C: VGPR v holds row M = rowBase + v (+8 for hi half), value |s_M|^2
        v8f Cm;
        #pragma unroll
        for (int v = 0; v < 8; ++v) Cm[v] = sn[rowBase + v + 8 * half];

        float rm[8];
        #pragma unroll
        for (int v = 0; v < 8; ++v) rm[v] = 3.0e38f;

        #pragma unroll
        for (int ct = 0; ct < 8; ++ct) {
            // D = A*B + C  == squared distances (full EXEC, uniform flow)
            v8f D = __builtin_amdgcn_wmma_f32_16x16x4_f32(
                false, A, false, Bt[ct], (short)0, Cm, false, false);

            // running row mins (d2)
            #pragma unroll
            for (int v = 0; v < 8; ++v) rm[v] = fminf(rm[v], D[v]);

            // column min over this tile's 8 rows (d2), tree
            float cmin = fminf(fminf(fminf(D[0], D[1]), fminf(D[2], D[3])),
                               fminf(fminf(D[4], D[5]), fminf(D[6], D[7])));
            // merge the two half-groups (different rows, same column)
            cmin = fminf(cmin, __shfl_xor(cmin, 16, 32));
            cmin = fmaxf(cmin, 0.0f);   // keep bit-ordering valid
            // merge across row tiles / waves (non-neg floats -> uint order OK)
            atomicMin(&colMinU[ct * 16 + lr], __float_as_uint(cmin));
        }

        // row mins: reduce d2 over the 16 lanes of each half-group (columns)
        #pragma unroll
        for (int v = 0; v < 8; ++v) {
            float r = rm[v];
            r = fminf(r, __shfl_xor(r, 1, 32));
            r = fminf(r, __shfl_xor(r, 2, 32));
            r = fminf(r, __shfl_xor(r, 4, 32));
            r = fminf(r, __shfl_xor(r, 8, 32));
            if (lr == 0) rowMinS[rowBase + v + 8 * half] = fast_sqrt(fmaxf(r, 0.0f));
        }
    }
    __syncthreads();

    // fitness = mean(rowmin) + mean(colmin); one sqrt per column here
    red[tid] = rowMinS[tid] + fast_sqrt(__uint_as_float(colMinU[tid]));
    __syncthreads();
    for (int s = 64; s > 0; s >>= 1) {
        if (tid < s) red[tid] += red[tid + s];
        __syncthreads();
    }
    if (tid == 0) fit[bp] = red[0] * (1.0f / (float)SSUB);
}

// ---------------- kernel 4: pbest / gbest update ----------------------------
__global__ __launch_bounds__(64) void best_update(
    const float* __restrict__ pos, const float* __restrict__ fit,
    float* __restrict__ pbp, float* __restrict__ pbf,
    float* __restrict__ gbp, float* __restrict__ gbf, int initFlag)
{
    __shared__ float v[64];
    __shared__ int   idx[64];
    int b = blockIdx.x, p = threadIdx.x;
    float f = 3.0e38f;
    if (p < NPART) {
        int i = b * NPART + p;
        f = fit[i];
        bool improve = initFlag ? true : (f < pbf[i]);
        if (improve) {
            pbf[i] = f;
            #pragma unroll
            for (int d = 0; d < 6; ++d) pbp[i * 6 + d] = pos[i * 6 + d];
        }
    }
    v[p] = f; idx[p] = p;
    __syncthreads();
    for (int s = 32; s > 0; s >>= 1) {
        if (p < s && v[p + s] < v[p]) { v[p] = v[p + s]; idx[p] = idx[p + s]; }
        __syncthreads();
    }
    if (p == 0) {
        float bf = v[0]; int bi = idx[0];
        bool improve = initFlag ? true : (bf < gbf[b]);
        if (improve) {
            gbf[b] = bf;
            #pragma unroll
            for (int d = 0; d < 6; ++d) gbp[b * 6 + d] = pos[(b * NPART + bi) * 6 + d];
        }
    }
}

// ---------------- kernel 5: PSO velocity/position update --------------------
__global__ void pso_update(
    float* __restrict__ pos, float* __restrict__ vel,
    const float* __restrict__ pbp, const float* __restrict__ gbp, int iter)
{
    int i = blockIdx.x * blockDim.x + threadIdx.x;
    if (i >= BATCH * NPART) return;
    int b = i / NPART;
    #pragma unroll
    for (int d = 0; d < 6; ++d) {
        unsigned base = (unsigned)iter * 1315423911u ^ (unsigned)i * 2654435761u ^ (unsigned)d * 40503u;
        float r1 = u01(base + 0x68BC21EBu);
        float r2 = u01(base + 0x02E5BE93u);
        float x = pos[i * 6 + d], v = vel[i * 6 + d];
        float pb = pbp[i * 6 + d], gb = gbp[b * 6 + d];
        v = 0.7f * v + 1.5f * r1 * (pb - x) + 1.5f * r2 * (gb - x);
        float vmax = (d < 3) ? ROT_RANGE * 0.2f : TRANS_RANGE * 0.2f;
        v = fminf(fmaxf(v, -vmax), vmax);
        float pmax = (d < 3) ? ROT_RANGE : TRANS_RANGE;
        x = fminf(fmaxf(x + v, -pmax), pmax);
        pos[i * 6 + d] = x; vel[i * 6 + d] = v;
    }
}

// ---------------- kernel 6: finalize R, t into d_out ------------------------
__global__ __launch_bounds__(64) void finalize_rt(const float* __restrict__ gbp,
                                                  float* __restrict__ out)
{
    int b = threadIdx.x;
    if (b >= BATCH) return;
    float R[9];
    rodrigues(gbp[b * 6 + 0], gbp[b * 6 + 1], gbp[b * 6 + 2], R);
    #pragma unroll
    for (int k = 0; k < 9; ++k) out[b * 9 + k] = R[k];
    #pragma unroll
    for (int k = 0; k < 3; ++k) out[BATCH * 9 + b * 3 + k] = gbp[b * 6 + 3 + k];
}

// ---------------- kernel 7: apply best transform to all points --------------
__global__ __launch_bounds__(256) void align_kernel(
    const float* __restrict__ src, const float* __restrict__ rt,
    float* __restrict__ out)
{
    __shared__ float R[9], T[3];
    const int blocksPerBatch = NPTS / 256;
    int b = blockIdx.x / blocksPerBatch;
    int n = (blockIdx.x % blocksPerBatch) * 256 + threadIdx.x;
    if (threadIdx.x < 9) R[threadIdx.x] = rt[b * 9 + threadIdx.x];
    if (threadIdx.x < 3) T[threadIdx.x] = rt[BATCH * 9 + b * 3 + threadIdx.x];
    __syncthreads();
    size_t gi = ((size_t)b * NPTS + n) * 3;
    float x = src[gi], y = src[gi + 1], z = src[gi + 2];
    out[gi + 0] = R[0] * x + R[1] * y + R[2] * z + T[0];
    out[gi + 1] = R[3] * x + R[4] * y + R[5] * z + T[1];
    out[gi + 2] = R[6] * x + R[7] * y + R[8] * z + T[2];
}

// ---------------- host-side orchestration -----------------------------------
extern "C" void kernel_launch(void* const* d_in, const int* in_sizes, int n_in,
                              void* d_out, int out_size, void* d_ws, size_t ws_size,
                              hipStream_t stream)
{
    const float* src = (const float*)d_in[0];
    const float* tgt = (const float*)d_in[1];
    float* out = (float*)d_out;

    // workspace carve-up (floats)
    float* w   = (float*)d_ws;
    float* ssx = w;
    float* ssy = ssx + BATCH * SSUB;
    float* ssz = ssy + BATCH * SSUB;
    float* ttx = ssz + BATCH * SSUB;
    float* tty = ttx + BATCH * SSUB;
    float* ttz = tty + BATCH * SSUB;
    float* ttn = ttz + BATCH * SSUB;
    float* pos = ttn + BATCH * SSUB;           // B*P*6
    float* vel = pos + BATCH * NPART * 6;
    float* pbp = vel + BATCH * NPART * 6;
    float* pbf = pbp + BATCH * NPART * 6;      // B*P
    float* fit = pbf + BATCH * NPART;          // B*P
    float* gbp = fit + BATCH * NPART;          // B*6
    float* gbf = gbp + BATCH * 6;              // B

    subsample_kernel<<<BATCH, SSUB, 0, stream>>>(src, tgt, ssx, ssy, ssz,
                                                 ttx, tty, ttz, ttn);
    init_particles<<<(BATCH * NPART + 127) / 128, 128, 0, stream>>>(pos, vel);

    eval_wmma_kernel<<<BATCH * NPART, 128, 0, stream>>>(ssx, ssy, ssz,
                                                        ttx, tty, ttz, ttn,
                                                        pos, fit);
    best_update<<<BATCH, 64, 0, stream>>>(pos, fit, pbp, pbf, gbp, gbf, 1);

    for (int it = 0; it < NITER; ++it) {
        pso_update<<<(BATCH * NPART + 127) / 128, 128, 0, stream>>>(pos, vel, pbp, gbp, it);
        eval_wmma_kernel<<<BATCH * NPART, 128, 0, stream>>>(ssx, ssy, ssz,
                                                            ttx, tty, ttz, ttn,
                                                            pos, fit);
        best_update<<<BATCH, 64, 0, stream>>>(pos, fit, pbp, pbf, gbp, gbf, 0);
    }

    finalize_rt<<<1, 64, 0, stream>>>(gbp, out);
    align_kernel<<<BATCH * (NPTS / 256), 256, 0, stream>>>(src, out,
                                                           out + BATCH * 9 + BATCH * 3);
}